// CrfDecodeLayer_47184510714344
// MI455X (gfx1250) — compile-verified
//
#include <hip/hip_runtime.h>

// Problem constants from the reference: B=128, T=2048, K=64.
#define B_ 128
#define T_ 2048
#define K_ 64
#define TILE 32
#define NT (T_ / TILE)

// ---- CDNA5 async-to-LDS helpers (ASYNCcnt path, CDNA5 ISA) ----
#define S_WAIT_ASYNCCNT(n) asm volatile("s_wait_asynccnt " #n ::: "memory")

__device__ __forceinline__ unsigned lds_addr_of(const void* p) {
  // Generic pointer to LDS: low 32 bits are the LDS byte offset (aperture layout).
  return (unsigned)(unsigned long long)p;
}
__device__ __forceinline__ void async_ld_b128(unsigned lds_byte_addr, const void* gaddr) {
  // GLOBAL_LOAD_ASYNC_TO_LDS_B128: VDST = LDS address VGPR, VADDR = 64-bit global address.
  asm volatile("global_load_async_to_lds_b128 %0, %1, off"
               :: "v"(lds_byte_addr), "v"(gaddr) : "memory");
}
__device__ __forceinline__ void async_ld_b32(unsigned lds_byte_addr, const void* gaddr) {
  asm volatile("global_load_async_to_lds_b32 %0, %1, off"
               :: "v"(lds_byte_addr), "v"(gaddr) : "memory");
}

// =====================  Forward: max-plus recurrence  =====================
// Grid: B blocks. Block: 128 threads = 4 wave32.
//   j = tid & 63  -> tag (output column)
//   h = tid >> 6  -> which half of the i-range [h*32, h*32+32) this thread reduces
// Transition column slice lives in 32 VGPRs per thread. Scores checkpointed to
// d_ws every step (L2-resident; 64 MiB) so the backtrack can recompute argmax.
__global__ __launch_bounds__(128) void viterbi_fwd(
    const float* __restrict__ emis, const int* __restrict__ mask,
    const float* __restrict__ trans, float* __restrict__ scores)
{
  __shared__ __align__(16) float sh_emis[2][TILE * K_];  // 2 x 8KB double buffer
  __shared__ int   sh_mask[2][TILE];
  __shared__ __align__(16) float sh_score[K_];
  __shared__ float sh_part[128];

  const int b   = blockIdx.x;
  const int tid = threadIdx.x;
  const int j   = tid & 63;
  const int h   = tid >> 6;

  // Per-thread transition column slice: tw[ii] = Tr[h*32+ii][j]  (coalesced over j)
  float tw[32];
#pragma unroll
  for (int ii = 0; ii < 32; ++ii)
    tw[ii] = trans[(h * 32 + ii) * K_ + j];

  const float* eb = emis   + (size_t)b * T_ * K_;
  const int*   mb = mask   + (size_t)b * T_;
  float*       sb = scores + (size_t)b * T_ * K_;

  // t = 0: score = emissions row 0
  if (h == 0) {
    float s0 = eb[j];
    sh_score[j] = s0;
    sb[j] = s0;
  }

  // Stage one emissions tile (8KB) + its mask slice into LDS, asynchronously.
  // Per wave: 4x async b128 + 1x async b32  -> ASYNCcnt +5 per tile.
  auto issue_tile = [&](int tile, int bufsel) {
    const float* src = eb + (size_t)tile * TILE * K_;
#pragma unroll
    for (int k = 0; k < 4; ++k) {
      int q = tid + 128 * k;  // 16-byte chunk index, consecutive lanes -> consecutive 16B
      async_ld_b128(lds_addr_of(&sh_emis[bufsel][q * 4]), src + q * 4);
    }
    int m = tid & 31;  // every wave stages the full mask slice (identical writes, harmless)
    async_ld_b32(lds_addr_of(&sh_mask[bufsel][m]), mb + tile * TILE + m);
  };

  issue_tile(0, 0);
  int buf = 0;

  for (int tile = 0; tile < NT; ++tile) {
    if (tile + 1 < NT) {
      issue_tile(tile + 1, buf ^ 1);   // 10 outstanding; wait until only next-tile's 5 remain
      S_WAIT_ASYNCCNT(5);
    } else {
      S_WAIT_ASYNCCNT(0);
    }
    __syncthreads();  // all waves' async data visible; also re-couples waves per tile

    const int t0     = tile * TILE;
    const int tstart = (tile == 0) ? 1 : 0;
    for (int tt = tstart; tt < TILE; ++tt) {
      const int t  = t0 + tt;
      const int mk = sh_mask[buf][tt];  // uniform across the block
      if (mk) {
        // Partial max over this thread's i-half: 8x ds_load_b128 + 32 add + 16 max3
        const float4* sv = (const float4*)(sh_score + h * 32);
        float m = -__builtin_inff();
#pragma unroll
        for (int q = 0; q < 8; ++q) {
          float4 s4 = sv[q];
          float c0 = s4.x + tw[4 * q + 0];
          float c1 = s4.y + tw[4 * q + 1];
          float c2 = s4.z + tw[4 * q + 2];
          float c3 = s4.w + tw[4 * q + 3];
          m = fmaxf(fmaxf(m, c0), c1);   // -> v_max3_num_f32
          m = fmaxf(fmaxf(m, c2), c3);
        }
        sh_part[tid] = m;
        __syncthreads();
        if (h == 0) {
          float best = fmaxf(sh_part[j], sh_part[64 + j]) + sh_emis[buf][tt * K_ + j];
          sh_score[j] = best;                 // only writer of sh_score[j]
          sb[(size_t)t * K_ + j] = best;      // checkpoint row (L2-resident)
        }
        __syncthreads();
      } else {
        // Masked step: score frozen; still checkpoint the row for the backtrack.
        if (h == 0) sb[(size_t)t * K_ + j] = sh_score[j];
      }
    }
    buf ^= 1;
  }
}

// =====================  Backtrack: recompute argmax  =====================
// Grid: B blocks, 1 wave32 per batch. Transposed transition in LDS so
// sh_trT[tag*64 + i] reads are bank-conflict-free across lanes. Argmax over the
// 64 candidates via shfl reduction with lowest-index tie-break (jnp.argmax).
__global__ __launch_bounds__(32) void viterbi_bwd(
    const float* __restrict__ scores, const int* __restrict__ mask,
    const float* __restrict__ trans, int* __restrict__ out)
{
  __shared__ float sh_trT[K_ * K_];  // 16KB
  const int b    = blockIdx.x;
  const int lane = threadIdx.x;

  for (int s = lane; s < K_ * K_; s += 32) {
    int i = s >> 6, jj = s & 63;
    sh_trT[jj * K_ + i] = trans[s];  // transpose on store
  }
  __syncthreads();

  const float* sb = scores + (size_t)b * T_ * K_;
  const int*   mb = mask   + (size_t)b * T_;
  int*         ob = out    + (size_t)b * T_;

  // last_tag = argmax(final score row), first-occurrence tie-break
  float v0 = sb[(size_t)(T_ - 1) * K_ + lane];
  float v1 = sb[(size_t)(T_ - 1) * K_ + lane + 32];
  int   i0 = lane, i1 = lane + 32;
  if (v1 > v0) { v0 = v1; i0 = i1; }
#pragma unroll
  for (int d = 16; d >= 1; d >>= 1) {
    float ov = __shfl_down(v0, d);
    int   oi = __shfl_down(i0, d);
    if (ov > v0 || (ov == v0 && oi < i0)) { v0 = ov; i0 = oi; }
  }
  int tag = __shfl(i0, 0);

  if (lane == 0) ob[T_ - 1] = mb[T_ - 1] ? tag : 0;

  // Software-pipelined: preload score row t-2 while reducing row t-1.
  float p0 = sb[(size_t)(T_ - 2) * K_ + lane];
  float p1 = sb[(size_t)(T_ - 2) * K_ + lane + 32];

  for (int t = T_ - 1; t >= 1; --t) {
    float r0 = p0, r1 = p1;
    if (t >= 2) {
      p0 = sb[(size_t)(t - 2) * K_ + lane];
      p1 = sb[(size_t)(t - 2) * K_ + lane + 32];
    }
    int mk = mb[t];  // uniform across wave
    if (mk) {
      float c0 = r0 + sh_trT[tag * K_ + lane];
      float c1 = r1 + sh_trT[tag * K_ + lane + 32];
      int   a0 = lane, a1 = lane + 32;
      if (c1 > c0) { c0 = c1; a0 = a1; }
#pragma unroll
      for (int d = 16; d >= 1; d >>= 1) {
        float ov = __shfl_down(c0, d);
        int   oi = __shfl_down(a0, d);
        if (ov > c0 || (ov == c0 && oi < a0)) { c0 = ov; a0 = oi; }
      }
      tag = __shfl(a0, 0);
    }
    if (lane == 0) ob[t - 1] = mb[t - 1] ? tag : 0;
  }
}

extern "C" void kernel_launch(void* const* d_in, const int* in_sizes, int n_in,
                              void* d_out, int out_size, void* d_ws, size_t ws_size,
                              hipStream_t stream) {
  (void)in_sizes; (void)n_in; (void)out_size; (void)ws_size;
  const float* emis  = (const float*)d_in[0];   // (B,T,K) f32
  const int*   mask  = (const int*)d_in[1];     // (B,T) i32
  const float* trans = (const float*)d_in[2];   // (K,K) f32
  float* scores = (float*)d_ws;                 // B*T*K f32 checkpoint = 64 MiB
  int*   out    = (int*)d_out;                  // (B,T) i32 tags

  viterbi_fwd<<<dim3(B_), dim3(128), 0, stream>>>(emis, mask, trans, scores);
  viterbi_bwd<<<dim3(B_), dim3(32),  0, stream>>>(scores, mask, trans, out);
}